// WavetableOsc_58308476010610
// MI455X (gfx1250) — compile-verified
//
#include <hip/hip_runtime.h>
#include <hip/hip_bf16.h>

// ---------------- problem constants ----------------
#define SRF      44100.0f
#define N_WT     2048
#define N_POS    64
#define AA_N     257            // filter taps
#define KPAD     260            // taps padded to multiple of 4 (zeros)
#define PADW     2320           // padded wavetable row width (>= 2048+259, mult of 16)
#define NS       262144         // samples per batch row
#define BS       32             // batch size
#define CHUNK    4096           // samples per scan chunk
#define NCHUNK   (NS / CHUNK)   // 64
#define TWO_PI_F 6.28318530717958647692f
#define PI_F     3.14159265358979323846f

typedef float v2f __attribute__((ext_vector_type(2)));
typedef float v8f __attribute__((ext_vector_type(8)));

#if __has_builtin(__builtin_amdgcn_global_load_async_to_lds_b32) && \
    __has_builtin(__builtin_amdgcn_s_wait_asynccnt)
#define USE_ASYNC_LDS 1
typedef __attribute__((address_space(1))) int g_as1_int;   // printed '__device__ int *'
typedef __attribute__((address_space(3))) int l_as3_int;   // LDS
#else
#define USE_ASYNC_LDS 0
#endif

// ---------------- workspace layout (float offsets) ----------------
#define WS_PADDED 0
#define WS_H      (WS_PADDED + N_POS * PADW)          // 64*2320   = 148480
#define WS_CSUM   (WS_H + BS * KPAD)                  // +32*260   =   8320
#define WS_CMAX   (WS_CSUM + BS * NCHUNK)             // +2048
#define WS_COFF   (WS_CMAX + BS * NCHUNK)             // +2048
#define WS_FILT   (WS_COFF + BS * NCHUNK)             // +2048
// filtered table: 32*64*2048 floats = 16 MB; total ws ~17.4 MB

// ============================================================
// 1) tanh + circular pad: padded[pos][j] = tanh(wt[pos][(j-128) mod 2048])
// ============================================================
__global__ __launch_bounds__(256) void k_prep_wt(const float* __restrict__ wt,
                                                 float* __restrict__ padded) {
  int idx = blockIdx.x * 256 + threadIdx.x;
  if (idx >= N_POS * PADW) return;
  int pos = idx / PADW;
  int j   = idx - pos * PADW;
  int src = (j + N_WT - (AA_N / 2)) % N_WT;   // j - 128, wrapped
  padded[idx] = tanhf(wt[pos * N_WT + src]);
}

// ============================================================
// 2) per-(batch, chunk) sum and max of f0
// ============================================================
__global__ __launch_bounds__(256) void k_partial(const float* __restrict__ f0,
                                                 float* __restrict__ csum,
                                                 float* __restrict__ cmax) {
  __shared__ float ssum[256];
  __shared__ float smax[256];
  const int c = blockIdx.x, b = blockIdx.y, t = threadIdx.x;
  const float* p = f0 + (size_t)b * NS + (size_t)c * CHUNK;
  float s = 0.0f, mx = -1e30f;
  for (int i = t; i < CHUNK; i += 256) {
    float v = p[i];
    s += v;
    mx = fmaxf(mx, v);
  }
  ssum[t] = s; smax[t] = mx;
  __syncthreads();
  for (int off = 128; off > 0; off >>= 1) {
    if (t < off) {
      ssum[t] += ssum[t + off];
      smax[t] = fmaxf(smax[t], smax[t + off]);
    }
    __syncthreads();
  }
  if (t == 0) {
    csum[b * NCHUNK + c] = ssum[0] * (TWO_PI_F / SRF);  // phase increment sum
    cmax[b * NCHUNK + c] = smax[0];
  }
}

// ============================================================
// 3) per-batch: max f0, chunk-offset exclusive scan, filter taps
// ============================================================
__global__ __launch_bounds__(320) void k_prep_h(const float* __restrict__ csum,
                                                const float* __restrict__ cmax,
                                                float* __restrict__ coff,
                                                float* __restrict__ H) {
  __shared__ float red[320];
  __shared__ float sMaxF0;
  __shared__ float sSum;
  const int b = blockIdx.x, t = threadIdx.x;
  if (t == 0) {
    float mx = -1e30f, run = 0.0f;
    for (int c = 0; c < NCHUNK; ++c) {
      mx = fmaxf(mx, cmax[b * NCHUNK + c]);
      coff[b * NCHUNK + c] = run;              // exclusive prefix of chunk phase sums
      run += csum[b * NCHUNK + c];
    }
    sMaxF0 = mx;
  }
  __syncthreads();
  const float maxf0 = sMaxF0;
  // cf = SR / (maxf0 / (SR/N_WT)) / 2 * 0.9
  const float cf = 0.45f * SRF * (SRF / (float)N_WT) / maxf0;
  float hk = 0.0f;
  if (t < AA_N) {
    float support = 2.0f * ((float)t - (float)(AA_N - 1) * 0.5f) / SRF;
    float x = cf * support;
    float snc = (fabsf(x) < 1e-8f) ? 1.0f : sinf(PI_F * x) / (PI_F * x);
    float ph = TWO_PI_F * (float)t / (float)(AA_N - 1);
    float bw = 0.42f - 0.5f * cosf(ph) + 0.08f * cosf(2.0f * ph);
    hk = snc * bw;
  }
  red[t] = hk;
  __syncthreads();
  if (t == 0) {
    float s = 0.0f;
    for (int i = 0; i < AA_N; ++i) s += red[i];
    sSum = s;
  }
  __syncthreads();
  if (t < KPAD) H[b * KPAD + t] = (t < AA_N) ? hk / sSum : 0.0f;
}

// ============================================================
// 4) WMMA GEMM: filtered[b, pos, w] = sum_k H[b,k] * padded[pos, w+k]
//    M = 32 (batch) -> two 16-row C tiles; N = 16 w's per wave; K = 260.
//    Block: 8 waves, one pos row, 128 w columns. LDS: H (33 KB) + row window,
//    filled via the CDNA5 async global->LDS path when available.
// ============================================================
__global__ __launch_bounds__(256) void k_filter_gemm(const float* __restrict__ padded,
                                                     const float* __restrict__ H,
                                                     float* __restrict__ filt) {
  __shared__ float sH[BS * KPAD];   // 8320 floats
  __shared__ float sRow[400];       // 128 + 260 = 388 used
  const int wbase = blockIdx.x * 128;  // 16 w-tiles of 8 waves
  const int pos   = blockIdx.y;
  const int t     = threadIdx.x;

#if USE_ASYNC_LDS
  for (int i = t; i < BS * KPAD; i += 256)
    __builtin_amdgcn_global_load_async_to_lds_b32(
        (g_as1_int*)&H[i], (l_as3_int*)&sH[i], 0, 0);
  for (int i = t; i < 388; i += 256)
    __builtin_amdgcn_global_load_async_to_lds_b32(
        (g_as1_int*)&padded[pos * PADW + wbase + i], (l_as3_int*)&sRow[i], 0, 0);
  __builtin_amdgcn_s_wait_asynccnt(0);
  __syncthreads();
#else
  for (int i = t; i < BS * KPAD; i += 256) sH[i] = H[i];
  for (int i = t; i < 388; i += 256) sRow[i] = padded[pos * PADW + wbase + i];
  __syncthreads();
#endif

  const int lane = t & 31;
  const int wave = t >> 5;
  const int l0   = wave * 16;       // this wave's w offset within the block
  const int hi   = lane >> 4;       // half-wave select
  const int m    = lane & 15;       // M (for A) and N (for B/C) lane index

  v8f c0 = {};                      // rows 0..15 (batch 0..15)
  v8f c1 = {};                      // rows 16..31

  const float* a0p = &sH[m * KPAD];
  const float* a1p = &sH[(m + 16) * KPAD];

  for (int k0 = 0; k0 < KPAD; k0 += 4) {
    const int ka = k0 + 2 * hi;     // A/B VGPR0 -> K = ka, VGPR1 -> K = ka+1
    v2f a0, a1, bb;
    a0.x = a0p[ka];            a0.y = a0p[ka + 1];
    a1.x = a1p[ka];            a1.y = a1p[ka + 1];
    bb.x = sRow[l0 + m + ka];  bb.y = sRow[l0 + m + ka + 1];
    c0 = __builtin_amdgcn_wmma_f32_16x16x4_f32(false, a0, false, bb,
                                               (short)0, c0, false, false);
    c1 = __builtin_amdgcn_wmma_f32_16x16x4_f32(false, a1, false, bb,
                                               (short)0, c1, false, false);
  }

  // C layout: VGPR v, lanes 0-15 -> M=v, lanes 16-31 -> M=v+8; N = lane%16
  float* outp = filt + (size_t)pos * N_WT + (size_t)(wbase + l0 + m);
  const size_t rowStride = (size_t)N_POS * N_WT;   // stride between batches
#pragma unroll
  for (int v = 0; v < 8; ++v) {
    const int r = v + 8 * hi;
    outp[(size_t)r * rowStride]        = c0[v];
    outp[(size_t)(r + 16) * rowStride] = c1[v];
  }
}

// ============================================================
// 5) phase scan + bilinear wavetable gather
// ============================================================
__global__ __launch_bounds__(256) void k_osc(const float* __restrict__ f0,
                                             const float* __restrict__ wtpos,
                                             const float* __restrict__ phase,
                                             const float* __restrict__ coff,
                                             const float* __restrict__ filt,
                                             float* __restrict__ out) {
  __shared__ float sscan[256];
  const int c = blockIdx.x, b = blockIdx.y, t = threadIdx.x;
  const size_t base_n = (size_t)b * NS + (size_t)c * CHUNK;
  const float* pf0 = f0 + base_n + (size_t)t * 16;
  const float* pwp = wtpos + base_n + (size_t)t * 16;

  // wt_pos isn't needed until after the block scan: prefetch it now so the
  // lines are in flight behind the scan barriers (global_prefetch_b8).
  __builtin_prefetch(pwp, 0, 3);

  float inc[16];
  float tsum = 0.0f;
#pragma unroll
  for (int i = 0; i < 16; ++i) {
    inc[i] = pf0[i] * (TWO_PI_F / SRF);
    tsum += inc[i];
  }
  sscan[t] = tsum;
  __syncthreads();
  for (int off = 1; off < 256; off <<= 1) {
    float v = sscan[t];
    float w = (t >= off) ? sscan[t - off] : 0.0f;
    __syncthreads();
    sscan[t] = v + w;
    __syncthreads();
  }
  const float excl = sscan[t] - tsum;

  float run = phase[b] + coff[b * NCHUNK + c] + excl;
  const float* F = filt + (size_t)b * (N_POS * N_WT);
  float* po = out + base_n + (size_t)t * 16;
  const float inv2pi = 1.0f / TWO_PI_F;

#pragma unroll 4
  for (int i = 0; i < 16; ++i) {
    run += inc[i];
    float arg = run - floorf(run * inv2pi) * TWO_PI_F;   // mod 2pi
    float x = arg * (1.0f / PI_F) - 1.0f;                // [-1,1) along wavetable
    float y = pwp[i];

    float xp = fminf(fmaxf((x + 1.0f) * 0.5f * (float)(N_WT - 1), 0.0f),
                     (float)(N_WT - 1));
    float yp = fminf(fmaxf((y + 1.0f) * 0.5f * (float)(N_POS - 1), 0.0f),
                     (float)(N_POS - 1));
    float x0f = floorf(xp), y0f = floorf(yp);
    float fx = xp - x0f, fy = yp - y0f;
    int x0 = (int)x0f, y0 = (int)y0f;
    int x1 = min(x0 + 1, N_WT - 1);
    int y1 = min(y0 + 1, N_POS - 1);

    const float* r0 = F + (size_t)y0 * N_WT;
    const float* r1 = F + (size_t)y1 * N_WT;
    float v00 = r0[x0], v01 = r0[x1];
    float v10 = r1[x0], v11 = r1[x1];
    float top = v00 + (v01 - v00) * fx;
    float bot = v10 + (v11 - v10) * fx;
    po[i] = top + (bot - top) * fy;
  }
}

// ============================================================
extern "C" void kernel_launch(void* const* d_in, const int* in_sizes, int n_in,
                              void* d_out, int out_size, void* d_ws, size_t ws_size,
                              hipStream_t stream) {
  const float* f0    = (const float*)d_in[0];   // [32, 262144]
  const float* wtpos = (const float*)d_in[1];   // [32, 262144]
  const float* phase = (const float*)d_in[2];   // [32, 1]
  const float* wt    = (const float*)d_in[3];   // [64, 2048]

  float* ws     = (float*)d_ws;
  float* padded = ws + WS_PADDED;
  float* H      = ws + WS_H;
  float* csum   = ws + WS_CSUM;
  float* cmax   = ws + WS_CMAX;
  float* coff   = ws + WS_COFF;
  float* filt   = ws + WS_FILT;

  (void)in_sizes; (void)n_in; (void)out_size; (void)ws_size;

  k_prep_wt<<<(N_POS * PADW + 255) / 256, 256, 0, stream>>>(wt, padded);
  k_partial<<<dim3(NCHUNK, BS), 256, 0, stream>>>(f0, csum, cmax);
  k_prep_h<<<BS, 320, 0, stream>>>(csum, cmax, coff, H);
  k_filter_gemm<<<dim3(16, N_POS), 256, 0, stream>>>(padded, H, filt);
  k_osc<<<dim3(NCHUNK, BS), 256, 0, stream>>>(f0, wtpos, phase, coff, filt,
                                              (float*)d_out);
}